// BiGNNLayer_51943334478077
// MI455X (gfx1250) — compile-verified
//
#include <hip/hip_runtime.h>

#define DIM 64

typedef float v2f __attribute__((ext_vector_type(2)));
typedef float v8f __attribute__((ext_vector_type(8)));

// ---------------------------------------------------------------------------
// Kernel 1: zero the SpMM accumulator (float4 stores, pure bandwidth)
// ---------------------------------------------------------------------------
__global__ void bignn_zero(float4* __restrict__ p, int n4) {
    int i = blockIdx.x * blockDim.x + threadIdx.x;
    if (i < n4) p[i] = make_float4(0.f, 0.f, 0.f, 0.f);
}

// ---------------------------------------------------------------------------
// Kernel 2: SpMM scatter.  One wave32 per edge; each lane handles 2 of the
// 64 feature columns.  edge_{row,col,val} loads are wave-uniform (scalarized),
// feature gathers hit the 192MB L2 (features = 25.6MB resident), accumulation
// via global_atomic_add_f32.  This kernel is the roofline bottleneck
// (~102M f32 atomics + 19MB edge stream from HBM).
// ---------------------------------------------------------------------------
__global__ void bignn_spmm(const int* __restrict__ erow,
                           const int* __restrict__ ecol,
                           const float* __restrict__ eval_,
                           const float* __restrict__ x,
                           float* __restrict__ y, int E) {
    int t = blockIdx.x * blockDim.x + threadIdx.x;
    int e = t >> 5;           // wave-uniform edge id
    if (e >= E) return;
    int lane = t & 31;
    int r = erow[e];
    int c = ecol[e];
    float v = eval_[e];
    int off = lane * 2;
    float2 xv = *(const float2*)(x + (long)c * DIM + off);
    float* dst = y + (long)r * DIM + off;
    atomicAdd(dst,     v * xv.x);
    atomicAdd(dst + 1, v * xv.y);
}

// ---------------------------------------------------------------------------
// Kernel 3: fused dual GEMM with fp32 WMMA:
//   out = (lap + feat) @ W1 + (lap * feat) @ W2 + (b1 + b2)
// Block = 256 threads = 8 waves; each wave computes a 16x64 output tile with
// V_WMMA_F32_16X16X4_F32.
//
// Weights are staged in LDS PRE-PAIRED along K:
//   sW[kp*64 + n] = { W[2*kp][n], W[2*kp+1][n] }   (kp = 0..31)
// so every B fragment is one aligned ds_load_b64 into an even VGPR pair —
// no cross-register shuffling between WMMAs.
//
// f32 WMMA fragment layouts (ISA 7.12.2):
//   A 16x4 : lane m=lane&15, VGPR{0,1} = K = k0*4 + (lane<16 ? 0 : 2) + {0,1}
//   B 4x16 : lane n=lane&15, same K split as A
//   C/D    : 8 VGPRs, row = g + (lane<16 ? 0 : 8), col = lane&15
//
// Tail handling: A rows >= N are loaded from a clamped (in-bounds) address;
// since D rows are independent, their garbage never reaches memory (stores
// are guarded).  EXEC stays all-1s through the WMMA loop.
// ---------------------------------------------------------------------------
__global__ __launch_bounds__(256)
void bignn_gemm(const float* __restrict__ lap,
                const float* __restrict__ feat,
                const float* __restrict__ W1, const float* __restrict__ b1,
                const float* __restrict__ W2, const float* __restrict__ b2,
                float* __restrict__ out, int N) {
    __shared__ v2f  sW1[(DIM / 2) * DIM];   // 16 KB, K-paired
    __shared__ v2f  sW2[(DIM / 2) * DIM];   // 16 KB, K-paired
    __shared__ float sBias[DIM];

    int tid = threadIdx.x;
    {
        float* w1f = (float*)sW1;
        float* w2f = (float*)sW2;
        for (int i = tid; i < DIM * DIM; i += 256) {
            int kRow = i >> 6;          // 0..63
            int n    = i & 63;
            int dst  = (kRow >> 1) * (2 * DIM) + n * 2 + (kRow & 1);
            w1f[dst] = W1[i];
            w2f[dst] = W2[i];
        }
        if (tid < DIM) sBias[tid] = b1[tid] + b2[tid];
    }
    __syncthreads();

    int wave = tid >> 5;
    int lane = tid & 31;
    int rowBase = blockIdx.x * 128 + wave * 16;
    if (rowBase >= N) return;   // whole-wave exit only; EXEC full otherwise

    int m    = lane & 15;            // A-row / B-col within tile
    int half = lane >> 4;            // 0: K+{0,1}, 1: K+{2,3}
    int row  = rowBase + m;
    int rowSafe = row < N ? row : (N - 1);   // clamp; garbage rows never stored

    const float2* lrow = (const float2*)(lap  + (long)rowSafe * DIM);
    const float2* frow = (const float2*)(feat + (long)rowSafe * DIM);

    v8f acc[4];
    acc[0] = v8f{}; acc[1] = v8f{}; acc[2] = v8f{}; acc[3] = v8f{};

    for (int k0 = 0; k0 < 16; ++k0) {
        int kp = k0 * 2 + half;          // K-pair index 0..31
        float2 l = lrow[kp];
        float2 f = frow[kp];
        v2f aS = { l.x + f.x, l.y + f.y };   // (lap + feat) fragment
        v2f aP = { l.x * f.x, l.y * f.y };   // (lap * feat) fragment
        #pragma unroll
        for (int nb = 0; nb < 4; ++nb) {
            int cidx = nb * 16 + m;
            v2f bA = sW1[kp * DIM + cidx];   // single ds_load_b64
            v2f bB = sW2[kp * DIM + cidx];
            acc[nb] = __builtin_amdgcn_wmma_f32_16x16x4_f32(
                false, aS, false, bA, (short)0, acc[nb], false, false);
            acc[nb] = __builtin_amdgcn_wmma_f32_16x16x4_f32(
                false, aP, false, bB, (short)0, acc[nb], false, false);
        }
    }

    int rOff = half * 8;             // C/D layout: lanes 16-31 hold rows M+8
    if (rowBase + 16 <= N) {
        // fast path: whole tile in range
        #pragma unroll
        for (int nb = 0; nb < 4; ++nb) {
            int col = nb * 16 + m;
            float bias = sBias[col];
            #pragma unroll
            for (int g = 0; g < 8; ++g) {
                out[(long)(rowBase + rOff + g) * DIM + col] = acc[nb][g] + bias;
            }
        }
    } else {
        #pragma unroll
        for (int nb = 0; nb < 4; ++nb) {
            int col = nb * 16 + m;
            float bias = sBias[col];
            #pragma unroll
            for (int g = 0; g < 8; ++g) {
                int r = rowBase + rOff + g;
                if (r < N) out[(long)r * DIM + col] = acc[nb][g] + bias;
            }
        }
    }
}

// ---------------------------------------------------------------------------
extern "C" void kernel_launch(void* const* d_in, const int* in_sizes, int n_in,
                              void* d_out, int out_size, void* d_ws, size_t ws_size,
                              hipStream_t stream) {
    const int*   erow  = (const int*)  d_in[0];
    const int*   ecol  = (const int*)  d_in[1];
    const float* eval_ = (const float*)d_in[2];
    const float* feat  = (const float*)d_in[3];
    const float* W1    = (const float*)d_in[4];
    const float* b1    = (const float*)d_in[5];
    const float* W2    = (const float*)d_in[6];
    const float* b2    = (const float*)d_in[7];
    float* out = (float*)d_out;
    float* lap = (float*)d_ws;        // needs N*64*4 = 25.6 MB of workspace

    int E = in_sizes[0];
    int N = in_sizes[3] / DIM;

    // 1) zero the accumulator
    int n4 = (N * DIM) / 4;
    bignn_zero<<<(n4 + 255) / 256, 256, 0, stream>>>((float4*)lap, n4);

    // 2) SpMM scatter: one wave per edge
    long threads = (long)E * 32;
    int blocks = (int)((threads + 255) / 256);
    bignn_spmm<<<blocks, 256, 0, stream>>>(erow, ecol, eval_, feat, lap, E);

    // 3) fused dual WMMA GEMM + biases
    bignn_gemm<<<(N + 127) / 128, 256, 0, stream>>>(lap, feat, W1, b1, W2, b2, out, N);
}